// BilateralFilterLayer_33062658245441
// MI455X (gfx1250) — compile-verified
//
#include <hip/hip_runtime.h>
#include <hip/hip_bf16.h>
#include <stdint.h>

// Bilateral 3x3 filter, sigma_space = sigma_color = 0.8, reflect padding.
//
// out = sum_ij( ws_ij * exp(-(p_ij-c)^2/(2*0.64)) * p_ij )
//     / sum_ij( ws_ij * exp(-(p_ij-c)^2/(2*0.64)) )
// (both normalizations in the reference cancel in this ratio)
//
// Memory-bound: ~50MB total traffic -> ~2.2us floor @ 23.3 TB/s.
// CDNA5 path used: global_load_async_to_lds_b32 (ASYNCcnt) to stage tiles+halo
// into LDS with reflect indexing folded into the per-lane copy addresses.

#define IMG_H 512
#define IMG_W 512
#define TW 64          // tile width  (one thread per column)
#define TH 16          // tile height (4 rows per thread, 256 threads)
#define LW (TW + 2)    // 66 (LDS row stride, with halo)
#define LH (TH + 2)    // 18

#define KCOLOR (-1.1271055f)   // -1/(2*0.8^2) * log2(e) = -0.78125 * 1.4426950
#define SW_E   0.45783327f     // exp(-0.78125)       (edge spatial weight)
#define SW_EE  0.20961113f     // exp(-0.78125)^2     (corner spatial weight)

__global__ __launch_bounds__(256) void bilateral3x3_kernel(
    const float* __restrict__ img, float* __restrict__ out) {
  __shared__ float tile[LH][LW];

  const int tx0   = blockIdx.x * TW;
  const int ty0   = blockIdx.y * TH;
  const int plane = blockIdx.z;                        // b*C + c
  const float* __restrict__ src = img + (size_t)plane * (IMG_H * IMG_W);
  float* __restrict__ dst       = out + (size_t)plane * (IMG_H * IMG_W);

  const int tid = threadIdx.x;

  // ---- async-stage tile + reflect halo into LDS (gfx1250 async path) ----
  const uint32_t ldsBase = (uint32_t)(uintptr_t)(&tile[0][0]);
  #pragma unroll
  for (int i = tid; i < LH * LW; i += 256) {
    int ly = i / LW;
    int lx = i - ly * LW;
    int gy = ty0 + ly - 1;
    int gx = tx0 + lx - 1;
    gy = (gy < 0) ? -gy : gy;                   // reflect (no edge repeat)
    gy = (gy >= IMG_H) ? (2 * IMG_H - 2 - gy) : gy;
    gx = (gx < 0) ? -gx : gx;
    gx = (gx >= IMG_W) ? (2 * IMG_W - 2 - gx) : gx;
    uint64_t gaddr = (uint64_t)(uintptr_t)(src + gy * IMG_W + gx);
    uint32_t laddr = ldsBase + (uint32_t)i * 4u;
    asm volatile("global_load_async_to_lds_b32 %0, %1, off"
                 :: "v"(laddr), "v"(gaddr) : "memory");
  }
  asm volatile("s_wait_asynccnt 0x0" ::: "memory");
  __syncthreads();

  // ---- compute: one column per thread, 4 rows, sliding 3x3 window ----
  const int x  = tid & (TW - 1);   // 0..63
  const int y0 = (tid >> 6) * 4;   // 0,4,8,12

  float a0 = tile[y0 + 0][x + 0], a1 = tile[y0 + 0][x + 1], a2 = tile[y0 + 0][x + 2];
  float b0 = tile[y0 + 1][x + 0], b1 = tile[y0 + 1][x + 1], b2 = tile[y0 + 1][x + 2];

  #pragma unroll
  for (int r = 0; r < 4; ++r) {
    const float c0 = tile[y0 + 2 + r][x + 0];
    const float c1 = tile[y0 + 2 + r][x + 1];
    const float c2 = tile[y0 + 2 + r][x + 2];
    const float c  = b1;                      // center pixel

    float S = 1.0f;                           // center: ws=1 (unnorm), wc=1
    float P = c;
    float d, w;
    d = a0 - c; w = SW_EE * __builtin_amdgcn_exp2f(KCOLOR * d * d); S += w; P = fmaf(w, a0, P);
    d = a1 - c; w = SW_E  * __builtin_amdgcn_exp2f(KCOLOR * d * d); S += w; P = fmaf(w, a1, P);
    d = a2 - c; w = SW_EE * __builtin_amdgcn_exp2f(KCOLOR * d * d); S += w; P = fmaf(w, a2, P);
    d = b0 - c; w = SW_E  * __builtin_amdgcn_exp2f(KCOLOR * d * d); S += w; P = fmaf(w, b0, P);
    d = b2 - c; w = SW_E  * __builtin_amdgcn_exp2f(KCOLOR * d * d); S += w; P = fmaf(w, b2, P);
    d = c0 - c; w = SW_EE * __builtin_amdgcn_exp2f(KCOLOR * d * d); S += w; P = fmaf(w, c0, P);
    d = c1 - c; w = SW_E  * __builtin_amdgcn_exp2f(KCOLOR * d * d); S += w; P = fmaf(w, c1, P);
    d = c2 - c; w = SW_EE * __builtin_amdgcn_exp2f(KCOLOR * d * d); S += w; P = fmaf(w, c2, P);

    dst[(size_t)(ty0 + y0 + r) * IMG_W + (tx0 + x)] = P * __builtin_amdgcn_rcpf(S);

    // slide window down one row
    a0 = b0; a1 = b1; a2 = b2;
    b0 = c0; b1 = c1; b2 = c2;
  }
}

extern "C" void kernel_launch(void* const* d_in, const int* in_sizes, int n_in,
                              void* d_out, int out_size, void* d_ws, size_t ws_size,
                              hipStream_t stream) {
  const float* img = (const float*)d_in[0];
  float* out       = (float*)d_out;
  const int planes = in_sizes[0] / (IMG_H * IMG_W);   // B*C = 24
  dim3 grid(IMG_W / TW, IMG_H / TH, planes);          // (8, 32, 24)
  bilateral3x3_kernel<<<grid, 256, 0, stream>>>(img, out);
}